// Mobile_Former_31078383354488
// MI455X (gfx1250) — compile-verified
//
#include <hip/hip_runtime.h>
#include <hip/hip_bf16.h>

// Shapes (fixed by the reference)
#define Nb 64
#define Cc 384
#define Ll 3136        // 56*56
#define Mm 16
#define Dd 384
#define SPLITS 14      // L split into 14 ranges of 224 keys (14 blocks of 16)
#define KBLKS 14
#define QPAD 388       // q / x / fusion LDS row stride (floats), 16B aligned rows
#define TPAD 20        // lf tile LDS row stride (floats): 80B rows -> 16B aligned, conflict-free

typedef float v2f __attribute__((ext_vector_type(2)));
typedef float v8f __attribute__((ext_vector_type(8)));

__device__ __forceinline__ v8f wmma_f32(v2f a, v2f b, v8f c) {
  // D = A(16x4 f32) * B(4x16 f32) + C(16x16 f32)
  return __builtin_amdgcn_wmma_f32_16x16x4_f32(false, a, false, b, (short)0, c,
                                               false, false);
}

// ---------------------------------------------------------------------------
// Kernel 1: q = x @ Wq^T + bq   (per batch; 4 waves x 6 col-tiles of 16)
// ---------------------------------------------------------------------------
__global__ __launch_bounds__(128) void qproj_kernel(
    const float* __restrict__ x, const float* __restrict__ Wq,
    const float* __restrict__ bq, float* __restrict__ qout) {
  const int n = blockIdx.x;
  __shared__ float xs[16 * QPAD];
  const float* xn = x + (size_t)n * (Mm * Dd);
  for (int i = threadIdx.x * 4; i < Mm * Dd; i += 128 * 4) {
    const float4 v = *(const float4*)&xn[i];
    *(float4*)&xs[(i / Dd) * QPAD + (i % Dd)] = v;
  }
  __syncthreads();

  const int lane = threadIdx.x & 31;
  const int wv = threadIdx.x >> 5;
  const int half = lane >> 4, l16 = lane & 15;

  for (int t = 0; t < 6; ++t) {
    const int c0 = (wv * 6 + t) * 16;
    const int co = c0 + l16;                 // output channel for this lane
    v8f acc;
#pragma unroll
    for (int j = 0; j < 8; ++j) acc[j] = 0.f;
    const float* wrow = Wq + (size_t)co * Dd + 2 * half;  // B[k,n]=Wq[co][d]
#pragma unroll 8
    for (int ch = 0; ch < Dd / 4; ++ch) {
      v2f a = *(const v2f*)&xs[l16 * QPAD + ch * 4 + 2 * half];
      v2f b = *(const v2f*)&wrow[ch * 4];
      acc = wmma_f32(a, b, acc);
    }
    const float bias = bq[co];
    float* qo = qout + (size_t)n * (Mm * Cc) + co;
#pragma unroll
    for (int j = 0; j < 8; ++j) qo[(j + 8 * half) * Cc] = acc[j] + bias;
  }
}

// ---------------------------------------------------------------------------
// Kernel 2: flash-attention partial over one L-split (2 waves, 192ch each)
// ---------------------------------------------------------------------------
__global__ __launch_bounds__(64) void attn_partial_kernel(
    const float* __restrict__ lf, const float* __restrict__ q,
    float* __restrict__ opart, float* __restrict__ stats) {
  const int n = blockIdx.x;
  const int sp = blockIdx.y;
  const int lane = threadIdx.x & 31;
  const int wv = threadIdx.x >> 5;       // 0 or 1
  const int half = lane >> 4, l16 = lane & 15;

  __shared__ float qs[16 * QPAD];        // q[n], row-padded
  __shared__ float tile[2][Cc * TPAD];   // lf tile [c][l], double buffered
  __shared__ float sbuf[2][16 * TPAD];   // per-wave partial scores
  __shared__ float pbuf[16 * TPAD];      // probabilities (C-layout -> A-layout hop)

  // stage q[n] (24KB) into LDS, coalesced
  const float* qn = q + (size_t)n * (Mm * Cc);
  for (int i = threadIdx.x * 4; i < Mm * Cc; i += 64 * 4) {
    const float4 v = *(const float4*)&qn[i];
    *(float4*)&qs[(i / Cc) * QPAD + (i % Cc)] = v;
  }

  const size_t baseN = (size_t)n * Cc * (size_t)Ll;
  const int crow = 192 * wv;             // this wave's channel range base
  const int l0base = sp * (KBLKS * 16);

  // async copy this wave's 192 rows x 16 keys (12KB) into LDS buffer `buf`
  auto load_tile = [&](int buf, int l0) {
    const int row0 = crow + (lane >> 2);
    const int part = (lane & 3) * 4;     // 4 floats (b128) per lane
    const float* gp = lf + baseN + (size_t)row0 * Ll + l0 + part;
    unsigned lo = (unsigned)(size_t)&tile[buf][row0 * TPAD + part];
#pragma unroll
    for (int i = 0; i < 24; ++i) {       // 8 rows per instruction
      asm volatile("global_load_async_to_lds_b128 %0, %1, off"
                   :: "v"(lo + (unsigned)(i * 8 * TPAD * 4)),
                      "v"((unsigned long long)(size_t)(gp + (size_t)i * 8 * Ll))
                   : "memory");
    }
  };

  load_tile(0, l0base);                  // prefetch first block
  __syncthreads();                       // qs ready

  float mrow[8], lrow[8];
  v8f acc[12];
#pragma unroll
  for (int j = 0; j < 8; ++j) { mrow[j] = -3.4e38f; lrow[j] = 0.f; }
#pragma unroll
  for (int t = 0; t < 12; ++t)
#pragma unroll
    for (int j = 0; j < 8; ++j) acc[t][j] = 0.f;

  for (int kb = 0; kb < KBLKS; ++kb) {
    const int buf = kb & 1;
    if (kb + 1 < KBLKS) {
      load_tile(buf ^ 1, l0base + (kb + 1) * 16);     // prefetch next
      asm volatile("s_wait_asynccnt 24" ::: "memory"); // current block done
    } else {
      asm volatile("s_wait_asynccnt 0" ::: "memory");
    }
    const float* tb = &tile[buf][0];

    // ---- partial scores over this wave's 192 channels: 48 f32 WMMAs ----
    v8f s;
#pragma unroll
    for (int j = 0; j < 8; ++j) s[j] = 0.f;
#pragma unroll 6
    for (int ch = 0; ch < 48; ++ch) {
      const int c0 = crow + ch * 4 + 2 * half;
      v2f a = *(const v2f*)&qs[l16 * QPAD + crow + ch * 4 + 2 * half];
      v2f b;
      b.x = tb[(c0) * TPAD + l16];
      b.y = tb[(c0 + 1) * TPAD + l16];
      s = wmma_f32(a, b, s);
    }
#pragma unroll
    for (int j = 0; j < 8; ++j) sbuf[wv][(j + 8 * half) * TPAD + l16] = s[j];
    __syncthreads();
#pragma unroll
    for (int j = 0; j < 8; ++j)
      s[j] = (sbuf[0][(j + 8 * half) * TPAD + l16] +
              sbuf[1][(j + 8 * half) * TPAD + l16]) * 0.051031036307982884f;

    // ---- online softmax (rows live across a 16-lane half) ----
#pragma unroll
    for (int j = 0; j < 8; ++j) {
      float r = s[j];
#pragma unroll
      for (int o = 8; o >= 1; o >>= 1) r = fmaxf(r, __shfl_xor(r, o, 16));
      const float mo = mrow[j];
      const float mn = fmaxf(mo, r);
      const float corr = __expf(mo - mn);
      mrow[j] = mn;
      const float p = __expf(s[j] - mn);
      s[j] = p;
      float rs = p;
#pragma unroll
      for (int o = 8; o >= 1; o >>= 1) rs += __shfl_xor(rs, o, 16);
      lrow[j] = lrow[j] * corr + rs;
#pragma unroll
      for (int t = 0; t < 12; ++t) acc[t][j] *= corr;
    }

    // ---- P: C-layout -> LDS -> A-layout fragments (identical in both waves)
#pragma unroll
    for (int j = 0; j < 8; ++j) pbuf[(j + 8 * half) * TPAD + l16] = s[j];
    __syncthreads();

    v2f pa[4];
#pragma unroll
    for (int kc = 0; kc < 4; ++kc)
      pa[kc] = *(const v2f*)&pbuf[l16 * TPAD + kc * 4 + 2 * half];

    // ---- acc += P(16x16) * V(16x192): 12 tiles x 4 WMMAs ----
#pragma unroll
    for (int t = 0; t < 12; ++t) {
      const int c0 = crow + t * 16 + l16;
#pragma unroll
      for (int kc = 0; kc < 4; ++kc) {
        v2f b = *(const v2f*)&tb[c0 * TPAD + kc * 4 + 2 * half];
        acc[t] = wmma_f32(pa[kc], b, acc[t]);
      }
    }
  }

  // ---- write split partial + row stats ----
  float* op = opart + ((size_t)(n * SPLITS + sp)) * (Mm * Cc);
#pragma unroll
  for (int t = 0; t < 12; ++t) {
    const int c0 = crow + t * 16 + l16;
#pragma unroll
    for (int j = 0; j < 8; ++j) op[(j + 8 * half) * Cc + c0] = acc[t][j];
  }
  if (wv == 0 && l16 < 8) {
    float mv = 0.f, lv = 0.f;
#pragma unroll
    for (int j = 0; j < 8; ++j)
      if (j == l16) { mv = mrow[j]; lv = lrow[j]; }
    float* st = stats + ((size_t)(n * SPLITS + sp)) * 32;
    st[half * 8 + l16] = mv;           // rowmax, rows 0..15
    st[16 + half * 8 + l16] = lv;      // rowsum
  }
}

// ---------------------------------------------------------------------------
// Kernel 3: LSE-combine splits, then out = fusion @ Wu^T + bu + x
// ---------------------------------------------------------------------------
__global__ __launch_bounds__(128) void combine_proj_kernel(
    const float* __restrict__ opart, const float* __restrict__ stats,
    const float* __restrict__ x, const float* __restrict__ Wu,
    const float* __restrict__ bu, float* __restrict__ out) {
  const int n = blockIdx.x;
  __shared__ float fus[16 * QPAD];
  __shared__ float Mrow[16], Lsum[16];
  __shared__ float wexp[SPLITS * 16];
  const int tid = threadIdx.x;
  const float* st = stats + (size_t)n * SPLITS * 32;

  if (tid < 16) {
    float mx = -3.4e38f;
    for (int s = 0; s < SPLITS; ++s) mx = fmaxf(mx, st[s * 32 + tid]);
    float ls = 0.f;
    for (int s = 0; s < SPLITS; ++s)
      ls += __expf(st[s * 32 + tid] - mx) * st[s * 32 + 16 + tid];
    Mrow[tid] = mx; Lsum[tid] = ls;
  }
  __syncthreads();
  for (int i = tid; i < SPLITS * 16; i += 128) {
    const int s = i >> 4, m = i & 15;
    wexp[i] = __expf(st[s * 32 + m] - Mrow[m]);
  }
  __syncthreads();

  const float* opn = opart + (size_t)n * SPLITS * (Mm * Cc);
  for (int i = tid; i < Mm * Cc; i += 128) {
    const int m = i / Cc, c = i % Cc;
    float a = 0.f;
    for (int s = 0; s < SPLITS; ++s) a += wexp[s * 16 + m] * opn[s * (Mm * Cc) + i];
    fus[m * QPAD + c] = a / Lsum[m];
  }
  __syncthreads();

  const int lane = tid & 31, wv = tid >> 5;
  const int half = lane >> 4, l16 = lane & 15;
  const float* xn = x + (size_t)n * (Mm * Dd);
  float* on = out + (size_t)n * (Mm * Dd);
  for (int t = 0; t < 6; ++t) {
    const int d0 = (wv * 6 + t) * 16;
    const int dd = d0 + l16;
    v8f acc;
#pragma unroll
    for (int j = 0; j < 8; ++j) acc[j] = 0.f;
    const float* wrow = Wu + (size_t)dd * Cc + 2 * half;  // B[k,n]=Wu[d][c]
#pragma unroll 8
    for (int ch = 0; ch < Cc / 4; ++ch) {
      v2f a = *(const v2f*)&fus[l16 * QPAD + ch * 4 + 2 * half];
      v2f b = *(const v2f*)&wrow[ch * 4];
      acc = wmma_f32(a, b, acc);
    }
    const float bias = bu[dd];
#pragma unroll
    for (int j = 0; j < 8; ++j) {
      const int idx = (j + 8 * half) * Dd + dd;
      on[idx] = acc[j] + bias + xn[idx];
    }
  }
}

// ---------------------------------------------------------------------------
extern "C" void kernel_launch(void* const* d_in, const int* in_sizes, int n_in,
                              void* d_out, int out_size, void* d_ws, size_t ws_size,
                              hipStream_t stream) {
  const float* lf = (const float*)d_in[0];
  const float* x  = (const float*)d_in[1];
  const float* Wq = (const float*)d_in[2];
  const float* bq = (const float*)d_in[3];
  const float* Wu = (const float*)d_in[4];
  const float* bu = (const float*)d_in[5];
  float* out = (float*)d_out;

  float* qws   = (float*)d_ws;                               // 64*16*384
  float* opart = qws + (size_t)Nb * Mm * Cc;                 // 64*14*16*384
  float* stats = opart + (size_t)Nb * SPLITS * Mm * Cc;      // 64*14*32

  qproj_kernel<<<Nb, 128, 0, stream>>>(x, Wq, bq, qws);
  attn_partial_kernel<<<dim3(Nb, SPLITS), 64, 0, stream>>>(lf, qws, opart, stats);
  combine_proj_kernel<<<Nb, 128, 0, stream>>>(opart, stats, x, Wu, bu, out);
}